// G2C_27350351741742
// MI455X (gfx1250) — compile-verified
//
#include <hip/hip_runtime.h>
#include <hip/hip_bf16.h>
#include <math.h>

// Problem constants (from reference): B=128, N=256, T=100
#define B_SZ 128
#define N_SZ 256
#define T_IT 100
#define EPS_C 0.1f
#define ALPHA_C 5.0f
#define ALPHA_BASE_C 0.1f
// mask = diag_mask with diagonal set to 1 -> all ones -> sum = N*N
#define MASK_SUM 65536.0f

typedef __attribute__((ext_vector_type(2))) float v2f;
typedef __attribute__((ext_vector_type(8))) float v8f;

// fast hardware ops (args are bounded away from denormals in this kernel)
__device__ __forceinline__ float fast_rsqrt(float x) {
  return __builtin_amdgcn_rsqf(x);
}
__device__ __forceinline__ float fast_sqrt(float x) {
  return __builtin_amdgcn_sqrtf(x);
}

__device__ __forceinline__ float softplus_f(float x) {
  return (x > 20.0f) ? x : log1pf(__expf(x));
}

__device__ __forceinline__ unsigned hash_u32(unsigned x) {
  x ^= x >> 16; x *= 0x7feb352dU;
  x ^= x >> 15; x *= 0x846ca68bU;
  x ^= x >> 16; return x;
}

// Deterministic hash-based standard normal (Box-Muller)
__device__ __forceinline__ float gauss_f(unsigned seed) {
  unsigned a = hash_u32(seed * 0x9E3779B9U + 0x85EBCA6BU);
  unsigned b = hash_u32(a ^ 0xC2B2AE35U);
  float u1 = ((float)(a >> 8) + 1.0f) * (1.0f / 16777217.0f);
  float u2 = (float)(b >> 8) * (1.0f / 16777216.0f);
  return fast_sqrt(-2.0f * __logf(u1)) * __cosf(6.28318530718f * u2);
}

// ---------------------------------------------------------------------------
// Kernel 1: interleaved D/W preprocessing (one b64 per pair downstream).
// DW[b,i,j] = { softplus(ep[...,0]+ep^T[...,0]), softplus(...,1)+...^T } * mask
// ---------------------------------------------------------------------------
__global__ void g2c_prep_DW(const float* __restrict__ ep,
                            const float* __restrict__ dmask,
                            float2* __restrict__ DW) {
  long long idx = (long long)blockIdx.x * blockDim.x + threadIdx.x;
  const long long total = (long long)B_SZ * N_SZ * N_SZ;
  if (idx >= total) return;
  int j = (int)(idx % N_SZ);
  long long t = idx / N_SZ;
  int i = (int)(t % N_SZ);
  int b = (int)(t / N_SZ);
  long long base = (long long)b * N_SZ * N_SZ;
  const float2* ep2 = (const float2*)ep;
  float2 eij = ep2[base + (long long)i * N_SZ + j];
  float2 eji = ep2[base + (long long)j * N_SZ + i];
  float m = dmask[idx];
  float2 dw;
  dw.x = softplus_f(eij.x + eji.x) * m;
  dw.y = softplus_f(eij.y + eji.y) * m;
  DW[idx] = dw;
}

// ---------------------------------------------------------------------------
// Kernel 2a: row sums of D^2 (D symmetric -> row sums == col sums)
// ---------------------------------------------------------------------------
__global__ void g2c_row_sums(const float2* __restrict__ DW, float* __restrict__ rs) {
  __shared__ float sm[256];
  float d = DW[(long long)blockIdx.x * N_SZ + threadIdx.x].x;
  sm[threadIdx.x] = d * d;
  __syncthreads();
  for (int s = 128; s > 0; s >>= 1) {
    if (threadIdx.x < s) sm[threadIdx.x] += sm[threadIdx.x + s];
    __syncthreads();
  }
  if (threadIdx.x == 0) rs[blockIdx.x] = sm[0];
}

// Kernel 2b: per-batch total of D^2
__global__ void g2c_tot_sums(const float* __restrict__ rs, float* __restrict__ tot) {
  __shared__ float sm[256];
  sm[threadIdx.x] = rs[blockIdx.x * N_SZ + threadIdx.x];
  __syncthreads();
  for (int s = 128; s > 0; s >>= 1) {
    if (threadIdx.x < s) sm[threadIdx.x] += sm[threadIdx.x + s];
    __syncthreads();
  }
  if (threadIdx.x == 0) tot[blockIdx.x] = sm[0];
}

// ---------------------------------------------------------------------------
// Kernel 3: rank-3 power iteration on B_gram (on the fly from D), x_init+noise.
// Gram[j][i] = -0.5*(D[j][i]^2 - rs[j] - rs[i] + tot)
// ---------------------------------------------------------------------------
__global__ void g2c_init_X(const float2* __restrict__ DW,
                           const float* __restrict__ rs,
                           const float* __restrict__ tot,
                           float* __restrict__ X) {
  __shared__ float rsl[N_SZ];
  __shared__ float ul[N_SZ];
  __shared__ float upv[3][N_SZ];
  __shared__ float red[256];
  __shared__ float dots[4];
  int b = blockIdx.x, i = threadIdx.x;
  const float2* DWb = DW + (long long)b * N_SZ * N_SZ;
  rsl[i] = rs[b * N_SZ + i];
  float tt = tot[b];
  __syncthreads();

  for (int k = 0; k < 3; k++) {
    ul[i] = gauss_f((unsigned)(b * 7919 + k * 104729 + i * 131 + 17));
    __syncthreads();
    for (int step = 0; step < 10; step++) {
      // normalize u (eps 1e-3)
      float u = ul[i];
      red[i] = u * u;
      __syncthreads();
      for (int s = 128; s > 0; s >>= 1) {
        if (i < s) red[i] += red[i + s];
        __syncthreads();
      }
      float nrmv = fast_sqrt(red[0]);
      u = u / fmaxf(nrmv, 0.001f);
      __syncthreads();
      ul[i] = u;
      __syncthreads();
      // deflation dots with previous eigenvectors
      for (int p = 0; p < k; p++) {
        red[i] = upv[p][i] * u;
        __syncthreads();
        for (int s = 128; s > 0; s >>= 1) {
          if (i < s) red[i] += red[i + s];
          __syncthreads();
        }
        if (i == 0) dots[p] = red[0];
        __syncthreads();
      }
      // v = A_lr @ u  (A symmetric -> scan columns for coalescing)
      float acc = 0.0f;
      float rsi = rsl[i];
      for (int j = 0; j < N_SZ; j++) {
        float d = DWb[(long long)j * N_SZ + i].x;
        float a = -0.5f * (d * d - rsl[j] - rsi + tt);
        acc = fmaf(a, ul[j], acc);
      }
      for (int p = 0; p < k; p++) acc -= upv[p][i] * dots[p];
      __syncthreads();
      ul[i] = acc;
      __syncthreads();
    }
    // u /= (eig_sq + 0.01)^0.25
    float u = ul[i];
    red[i] = u * u;
    __syncthreads();
    for (int s = 128; s > 0; s >>= 1) {
      if (i < s) red[i] += red[i + s];
      __syncthreads();
    }
    u = u * fast_rsqrt(fast_sqrt(red[0] + 0.01f));
    upv[k][i] = u;
    __syncthreads();
  }

  float4 xv;
  xv.x = upv[0][i] + gauss_f((unsigned)(b * 15485863 + i * 37 + 0 + 3));
  xv.y = upv[1][i] + gauss_f((unsigned)(b * 15485863 + i * 37 + 1 + 3));
  xv.z = upv[2][i] + gauss_f((unsigned)(b * 15485863 + i * 37 + 2 + 3));
  xv.w = 1.0f;
  ((float4*)X)[b * N_SZ + i] = xv;
}

// ---------------------------------------------------------------------------
// Kernel 4: the 100 NLSQ steps. One block (8 wave32) per batch.
// Pair matrix tiled 16x16; Gram tile + gradient reduction both via
// V_WMMA_F32_16X16X4_F32. X double-buffered in LDS, rows padded to
// {x,y,z,1,0} so all WMMA operand fetches are unconditional ds_load_b32
// (no EXEC save/restore in the hot loop).
//
// dx_k = (4*EPS/M) * (x_k * S_k - P_k),
//   coeff_kj = W_kj*(D_kj*rsqrt(|x_k-x_j|^2+0.01) - 1)
//   P = coeff @ X (WMMA cols 0..2), S = coeff @ 1 (WMMA col 3)
// ---------------------------------------------------------------------------
__global__ void __launch_bounds__(256, 1)
g2c_nlsq_iter(const float2* __restrict__ DW, float* __restrict__ X) {
  __shared__ float Xb[2][N_SZ][5];     // {x,y,z,1,0} double-buffered
  __shared__ float nsq[2][N_SZ];       // |x|^2 per point
  __shared__ float gbuf[N_SZ][4];      // per-row P.xyz, S
  __shared__ float ctile[8][16][17];   // per-wave coeff tile (C->A relayout)

  const int b = blockIdx.x;
  const int tid = threadIdx.x;
  const int lane = tid & 31;
  const int wv = tid >> 5;
  const int hi = (lane >= 16) ? 1 : 0;
  const int ln = lane & 15;
  const int ca = hi ? 2 : 0;   // A/B .x component index (k or k+2)
  const int cb = hi ? 4 : 1;   // A/B .y component index (zero-pad for hi)
  const int lc = (ln < 4) ? ln : 4;  // grad-B column: x,y,z,1, then zero-pad

  const float2* DWb = DW + (long long)b * N_SZ * N_SZ;

  float4 x0 = ((const float4*)X)[b * N_SZ + tid];
  Xb[0][tid][0] = x0.x; Xb[0][tid][1] = x0.y; Xb[0][tid][2] = x0.z;
  Xb[0][tid][3] = 1.0f; Xb[0][tid][4] = 0.0f;
  Xb[1][tid][3] = 1.0f; Xb[1][tid][4] = 0.0f;
  nsq[0][tid] = x0.x * x0.x + x0.y * x0.y + x0.z * x0.z;
  __syncthreads();

  for (int t = 0; t < T_IT; t++) {
    const int cur = t & 1, nxt = cur ^ 1;
    float (*Xc)[5] = Xb[cur];
    float* nc = nsq[cur];

    // wave wv owns rows [wv*32, wv*32+32): two 16-row tiles
    for (int rt = 0; rt < 2; rt++) {
      const int i0 = (wv * 2 + rt) * 16;
      // A (Gram): A[m][k] = x_{i0+m}[k], K=3 padded to 4 (k3 = pad 0)
      v2f ga;
      ga.x = Xc[i0 + ln][ca];
      ga.y = Xc[i0 + ln][cb];
      // hoist |x_i|^2 for this row tile (invariant over j tiles)
      float niv[8];
#pragma unroll
      for (int v = 0; v < 8; v++) niv[v] = nc[i0 + hi * 8 + v];

      v8f acc = {};  // gradient accumulator (cols 0..2 = P, col 3 = S)

      for (int tj = 0; tj < 16; tj++) {
        const int j0 = tj * 16;
        // B (Gram): B[k][n] = x_{j0+n}[k]
        v2f gb;
        gb.x = Xc[j0 + ln][ca];
        gb.y = Xc[j0 + ln][cb];
        v8f z8 = {};
        v8f g = __builtin_amdgcn_wmma_f32_16x16x4_f32(
            false, ga, false, gb, (short)0, z8, false, false);

        // elementwise coeff on C/D layout: row = i0 + v + 8*hi, col = j0 + ln
        const float njv = nc[j0 + ln];
        const float2* dwrow =
            DWb + (long long)(i0 + hi * 8) * N_SZ + (j0 + ln);
        __builtin_prefetch(dwrow + 16, 0, 1);  // next j-tile (L2 resident)
#pragma unroll
        for (int v = 0; v < 8; v++) {
          float2 dw = dwrow[(long long)v * N_SZ];
          float r2 = niv[v] + njv - 2.0f * g[v] + 0.01f;  // >= 0.01
          float invd = fast_rsqrt(r2);                    // bare v_rsq_f32
          float c = dw.y * fmaf(dw.x, invd, -1.0f);
          ctile[wv][v + hi * 8][ln] = c;  // same-wave DS ops are ordered
        }

        // gradient: acc += coeff(16x16) @ [Xj | 1 | 0...], 4 K-chunks of 4
#pragma unroll
        for (int kc = 0; kc < 4; kc++) {
          v2f a2, b2;
          const int c0 = kc * 4 + hi * 2;
          a2.x = ctile[wv][ln][c0];
          a2.y = ctile[wv][ln][c0 + 1];
          const int k0 = j0 + kc * 4 + hi * 2;
          b2.x = Xc[k0][lc];       // cols: x,y,z,1,0-pad
          b2.y = Xc[k0 + 1][lc];
          acc = __builtin_amdgcn_wmma_f32_16x16x4_f32(
              false, a2, false, b2, (short)0, acc, false, false);
        }
      }
      // spill needed accumulator columns (0..3) to this wave's rows
      if (ln < 4) {
#pragma unroll
        for (int v = 0; v < 8; v++) gbuf[i0 + hi * 8 + v][ln] = acc[v];
      }
    }

    // per-row update: lane handles row r = wv*32 + lane (same-wave data only)
    const int r = wv * 32 + lane;
    const float S  = gbuf[r][3];
    const float px = gbuf[r][0], py = gbuf[r][1], pz = gbuf[r][2];
    float xx = Xc[r][0], xy = Xc[r][1], xz = Xc[r][2];
    const float Cg = 4.0f * EPS_C / MASK_SUM;
    float dx = Cg * (xx * S - px);
    float dy = Cg * (xy * S - py);
    float dz = Cg * (xz * S - pz);
    float speed = fast_sqrt(dx * dx + dy * dy + dz * dz + 0.001f);
    float alpha_t = ALPHA_BASE_C +
                    (ALPHA_C - ALPHA_BASE_C) * ((float)(T_IT - t) / (float)T_IT);
    float scale = alpha_t * tanhf(speed / alpha_t) / speed;
    xx += dx * scale; xy += dy * scale; xz += dz * scale;
    Xb[nxt][r][0] = xx; Xb[nxt][r][1] = xy; Xb[nxt][r][2] = xz;
    Xb[nxt][r][3] = 1.0f;
    nsq[nxt][r] = xx * xx + xy * xy + xz * xz;
    __syncthreads();
  }

  const int fin = T_IT & 1;
  float4 xo;
  xo.x = Xb[fin][tid][0]; xo.y = Xb[fin][tid][1]; xo.z = Xb[fin][tid][2];
  xo.w = 1.0f;
  ((float4*)X)[b * N_SZ + tid] = xo;
}

// ---------------------------------------------------------------------------
// Kernel 5: outputs. out0 = diag_mask * distances(X); out1 = diag_mask.
// ---------------------------------------------------------------------------
__global__ void g2c_outputs(const float* __restrict__ X,
                            const float* __restrict__ dmask,
                            float* __restrict__ out0, float* __restrict__ out1) {
  int b = blockIdx.x / N_SZ, i = blockIdx.x % N_SZ;
  __shared__ float4 xi_s;
  if (threadIdx.x == 0) xi_s = ((const float4*)X)[b * N_SZ + i];
  __syncthreads();
  float4 xi = xi_s;
  float4 xj = ((const float4*)X)[b * N_SZ + threadIdx.x];
  float ddx = xi.x - xj.x, ddy = xi.y - xj.y, ddz = xi.z - xj.z;
  float d = fast_sqrt(ddx * ddx + ddy * ddy + ddz * ddz + 0.01f);
  long long idx = ((long long)b * N_SZ + i) * N_SZ + threadIdx.x;
  float m = dmask[idx];
  out0[idx] = m * d;
  out1[idx] = m;
}

// ---------------------------------------------------------------------------
extern "C" void kernel_launch(void* const* d_in, const int* in_sizes, int n_in,
                              void* d_out, int out_size, void* d_ws, size_t ws_size,
                              hipStream_t stream) {
  (void)in_sizes; (void)n_in; (void)out_size; (void)ws_size;
  const float* ep = (const float*)d_in[0];     // [B,N,N,2]
  const float* dmask = (const float*)d_in[1];  // [B,N,N]

  const long long NN = (long long)B_SZ * N_SZ * N_SZ;  // 8,388,608
  float* ws = (float*)d_ws;
  float2* DW = (float2*)ws;                        // NN float2 (D,W interleaved)
  float* rs  = ws + 2 * NN;                        // B*N floats
  float* tot = rs + (long long)B_SZ * N_SZ;        // B floats
  float* X   = tot + B_SZ;                         // B*N*4 floats (16B aligned)

  g2c_prep_DW<<<(unsigned)((NN + 255) / 256), 256, 0, stream>>>(ep, dmask, DW);
  g2c_row_sums<<<B_SZ * N_SZ, 256, 0, stream>>>(DW, rs);
  g2c_tot_sums<<<B_SZ, 256, 0, stream>>>(rs, tot);
  g2c_init_X<<<B_SZ, 256, 0, stream>>>(DW, rs, tot, X);
  g2c_nlsq_iter<<<B_SZ, 256, 0, stream>>>(DW, X);

  float* out0 = (float*)d_out;
  float* out1 = out0 + NN;
  g2c_outputs<<<B_SZ * N_SZ, 256, 0, stream>>>(X, dmask, out0, out1);
}